// GraphSAGE_22591527976999
// MI455X (gfx1250) — compile-verified
//
#include <hip/hip_runtime.h>

typedef float v2f __attribute__((ext_vector_type(2)));
typedef float v8f __attribute__((ext_vector_type(8)));

#define N_NODES 100000
#define DIM 128
#define EPS 1e-5f

// ---------------------------------------------------------------------------
// utility: zero a float buffer
// ---------------------------------------------------------------------------
__global__ void k_zero(float* __restrict__ p, long long n) {
  long long i = (long long)blockIdx.x * blockDim.x + threadIdx.x;
  if (i < n) p[i] = 0.0f;
}

// ---------------------------------------------------------------------------
// degree: cnt[dst] += 1  (hardware f32 atomics)
// ---------------------------------------------------------------------------
__global__ void k_degree(const long long* __restrict__ dst, float* __restrict__ cnt,
                         long long E) {
  long long e = (long long)blockIdx.x * blockDim.x + threadIdx.x;
  if (e < E) unsafeAtomicAdd(&cnt[dst[e]], 1.0f);
}

// ---------------------------------------------------------------------------
// scatter: msg[dst] += h[src], vectorized float4 per thread (32 thr/edge)
// ---------------------------------------------------------------------------
__global__ void k_scatter(const float* __restrict__ h, const long long* __restrict__ src,
                          const long long* __restrict__ dst, float* __restrict__ msg,
                          long long E) {
  long long gid = (long long)blockIdx.x * blockDim.x + threadIdx.x;
  if (gid >= E * 32) return;
  long long e = gid >> 5;
  int c = (int)(gid & 31) * 4;
  long long s = src[e];
  long long d = dst[e];
  const float4 v = *(const float4*)(h + s * DIM + c);
  float* m = msg + d * DIM + c;
  unsafeAtomicAdd(m + 0, v.x);
  unsafeAtomicAdd(m + 1, v.y);
  unsafeAtomicAdd(m + 2, v.z);
  unsafeAtomicAdd(m + 3, v.w);
}

// ---------------------------------------------------------------------------
// Fused SAGE linear: out = (msg/max(cnt,1)) @ Wl^T + h @ Wr^T + b
// K = 128 fixed; H = 128 or 64.  WMMA f32 16x16x4, wave32.
// Block: 256 threads (8 waves), 32 nodes/block (N % 32 == 0).
// ---------------------------------------------------------------------------
template <int H>
__global__ __launch_bounds__(256) void k_sage_gemm(
    const float* __restrict__ msg, const float* __restrict__ cnt,
    const float* __restrict__ h, const float* __restrict__ Wl,
    const float* __restrict__ Wr, const float* __restrict__ bias,
    float* __restrict__ out) {
  constexpr int K = DIM;
  constexpr int PAD = 4;                     // row stride 132 -> conflict-free lanes
  __shared__ float sMean[32][K + PAD];
  __shared__ float sH[32][K + PAD];

  const int nodeBase = blockIdx.x * 32;
  const int tid = threadIdx.x;

  // cooperative staging: 32 rows x 32 float4 chunks, coalesced
  for (int i = tid; i < 32 * (K / 4); i += 256) {
    const int r  = i / (K / 4);
    const int c4 = (i % (K / 4)) * 4;
    const int node = nodeBase + r;
    const float inv = 1.0f / fmaxf(cnt[node], 1.0f);
    const float4 m4 = *(const float4*)(msg + (size_t)node * K + c4);
    const float4 h4 = *(const float4*)(h + (size_t)node * K + c4);
    sMean[r][c4 + 0] = m4.x * inv;
    sMean[r][c4 + 1] = m4.y * inv;
    sMean[r][c4 + 2] = m4.z * inv;
    sMean[r][c4 + 3] = m4.w * inv;
    sH[r][c4 + 0] = h4.x;
    sH[r][c4 + 1] = h4.y;
    sH[r][c4 + 2] = h4.z;
    sH[r][c4 + 3] = h4.w;
  }
  __syncthreads();

  const int wave = tid >> 5;
  const int lane = tid & 31;
  const int n  = lane & 15;                  // A row / B col / D col for this lane
  const int kk = (lane < 16) ? 0 : 2;        // K sub-offset per half-wave

  for (int ct = wave; ct < H / 16; ct += 8) {   // wave-uniform -> EXEC all 1s
    const int colBase = ct * 16;
    const float* wlRow = Wl + (size_t)(colBase + n) * K;
    const float* wrRow = Wr + (size_t)(colBase + n) * K;
    v8f c0 = {};                             // rows nodeBase .. +15
    v8f c1 = {};                             // rows nodeBase+16 .. +31

    for (int k = 0; k < K; k += 4) {
      const v2f bl  = *(const v2f*)(wlRow + k + kk);
      const v2f br  = *(const v2f*)(wrRow + k + kk);
      const v2f a0m = *(const v2f*)(&sMean[n][k + kk]);
      const v2f a0h = *(const v2f*)(&sH[n][k + kk]);
      const v2f a1m = *(const v2f*)(&sMean[16 + n][k + kk]);
      const v2f a1h = *(const v2f*)(&sH[16 + n][k + kk]);
      c0 = __builtin_amdgcn_wmma_f32_16x16x4_f32(false, a0m, false, bl, (short)0, c0, false, false);
      c0 = __builtin_amdgcn_wmma_f32_16x16x4_f32(false, a0h, false, br, (short)0, c0, false, false);
      c1 = __builtin_amdgcn_wmma_f32_16x16x4_f32(false, a1m, false, bl, (short)0, c1, false, false);
      c1 = __builtin_amdgcn_wmma_f32_16x16x4_f32(false, a1h, false, br, (short)0, c1, false, false);
    }

    const float bv = bias[colBase + n];
    const int mhalf = (lane < 16) ? 0 : 8;
#pragma unroll
    for (int j = 0; j < 8; ++j) {
      const int m = j + mhalf;
      out[(size_t)(nodeBase + m) * H + colBase + n]      = c0[j] + bv;
      out[(size_t)(nodeBase + 16 + m) * H + colBase + n] = c1[j] + bv;
    }
  }
}

// ---------------------------------------------------------------------------
// BatchNorm stats: stats[0:128]=sum, stats[128:256]=sumsq (column-parallel)
// ---------------------------------------------------------------------------
__global__ __launch_bounds__(128) void k_bn_stats(const float* __restrict__ x,
                                                  float* __restrict__ stats, int N) {
  const int col = threadIdx.x;               // 128 columns
  float s = 0.0f, s2 = 0.0f;
  for (int r = blockIdx.x; r < N; r += gridDim.x) {
    const float v = x[(size_t)r * DIM + col];
    s += v;
    s2 += v * v;
  }
  unsafeAtomicAdd(&stats[col], s);
  unsafeAtomicAdd(&stats[DIM + col], s2);
}

// ---------------------------------------------------------------------------
// BatchNorm affine + ReLU, in place
// ---------------------------------------------------------------------------
__global__ void k_bn_relu(float* __restrict__ x, const float* __restrict__ stats,
                          const float* __restrict__ g, const float* __restrict__ be,
                          int N) {
  const long long idx = (long long)blockIdx.x * blockDim.x + threadIdx.x;
  if (idx >= (long long)N * DIM) return;
  const int col = (int)(idx & (DIM - 1));
  const float invN = 1.0f / (float)N;
  const float mu = stats[col] * invN;
  const float var = stats[DIM + col] * invN - mu * mu;
  const float y = (x[idx] - mu) * rsqrtf(var + EPS) * g[col] + be[col];
  x[idx] = fmaxf(y, 0.0f);
}

// ---------------------------------------------------------------------------
// log_softmax over rows of 64, one wave32 per row (2 cols/lane)
// ---------------------------------------------------------------------------
__global__ __launch_bounds__(256) void k_logsoftmax64(float* __restrict__ out, int N) {
  const int row = blockIdx.x * 8 + (threadIdx.x >> 5);
  const int lane = threadIdx.x & 31;
  if (row >= N) return;
  float* p = out + (size_t)row * 64;
  const float v0 = p[lane];
  const float v1 = p[lane + 32];
  float m = fmaxf(v0, v1);
#pragma unroll
  for (int off = 16; off > 0; off >>= 1) m = fmaxf(m, __shfl_xor(m, off, 32));
  float s = expf(v0 - m) + expf(v1 - m);
#pragma unroll
  for (int off = 16; off > 0; off >>= 1) s += __shfl_xor(s, off, 32);
  const float lse = m + logf(s);
  p[lane] = v0 - lse;
  p[lane + 32] = v1 - lse;
}

// ---------------------------------------------------------------------------
// orchestration
// ---------------------------------------------------------------------------
extern "C" void kernel_launch(void* const* d_in, const int* in_sizes, int n_in,
                              void* d_out, int out_size, void* d_ws, size_t ws_size,
                              hipStream_t stream) {
  const float* x        = (const float*)d_in[0];
  const long long* ei   = (const long long*)d_in[1];
  const float* W1l = (const float*)d_in[2];
  const float* b1l = (const float*)d_in[3];
  const float* W1r = (const float*)d_in[4];
  const float* g1  = (const float*)d_in[5];
  const float* be1 = (const float*)d_in[6];
  const float* W2l = (const float*)d_in[7];
  const float* b2l = (const float*)d_in[8];
  const float* W2r = (const float*)d_in[9];
  const float* g2  = (const float*)d_in[10];
  const float* be2 = (const float*)d_in[11];
  const float* W3l = (const float*)d_in[12];
  const float* b3l = (const float*)d_in[13];
  const float* W3r = (const float*)d_in[14];

  const long long E = (long long)in_sizes[1] / 2;
  const long long N = N_NODES;
  const long long ND = N * DIM;
  const long long E32 = E * 32;

  const long long* src = ei;
  const long long* dst = ei + E;

  // workspace carve-up (floats)
  float* ws    = (float*)d_ws;
  float* cnt   = ws;                 // N
  float* msg   = cnt + N;            // N*128
  float* h1    = msg + ND;           // N*128
  float* h2    = h1 + ND;            // N*128
  float* stats = h2 + ND;            // 256
  float* outf  = (float*)d_out;      // N*64

  const int T = 256;
  const dim3 gZeroND((unsigned)((ND + T - 1) / T));
  const dim3 gZeroN((unsigned)((N + T - 1) / T));
  const dim3 gEdges((unsigned)((E + T - 1) / T));
  const dim3 gScat((unsigned)((E32 + T - 1) / T));
  const dim3 gGemm((unsigned)(N / 32));
  const dim3 gElem((unsigned)((ND + T - 1) / T));
  const dim3 gLsm((unsigned)((N + 7) / 8));

  // degree counts (shared by all 3 layers)
  k_zero<<<gZeroN, T, 0, stream>>>(cnt, N);
  k_degree<<<gEdges, T, 0, stream>>>(dst, cnt, E);

  // ---- layer 1 ----
  k_zero<<<gZeroND, T, 0, stream>>>(msg, ND);
  k_scatter<<<gScat, T, 0, stream>>>(x, src, dst, msg, E);
  k_sage_gemm<128><<<gGemm, T, 0, stream>>>(msg, cnt, x, W1l, W1r, b1l, h1);
  k_zero<<<dim3(1), T, 0, stream>>>(stats, 256);
  k_bn_stats<<<dim3(512), 128, 0, stream>>>(h1, stats, (int)N);
  k_bn_relu<<<gElem, T, 0, stream>>>(h1, stats, g1, be1, (int)N);

  // ---- layer 2 ----
  k_zero<<<gZeroND, T, 0, stream>>>(msg, ND);
  k_scatter<<<gScat, T, 0, stream>>>(h1, src, dst, msg, E);
  k_sage_gemm<128><<<gGemm, T, 0, stream>>>(msg, cnt, h1, W2l, W2r, b2l, h2);
  k_zero<<<dim3(1), T, 0, stream>>>(stats, 256);
  k_bn_stats<<<dim3(512), 128, 0, stream>>>(h2, stats, (int)N);
  k_bn_relu<<<gElem, T, 0, stream>>>(h2, stats, g2, be2, (int)N);

  // ---- layer 3 + log_softmax ----
  k_zero<<<gZeroND, T, 0, stream>>>(msg, ND);
  k_scatter<<<gScat, T, 0, stream>>>(h2, src, dst, msg, E);
  k_sage_gemm<64><<<gGemm, T, 0, stream>>>(msg, cnt, h2, W3l, W3r, b3l, outf);
  k_logsoftmax64<<<gLsm, T, 0, stream>>>(outf, (int)N);
}